// GCNN_ResidualLayer_64716567216741
// MI455X (gfx1250) — compile-verified
//
#include <hip/hip_runtime.h>

// ---------------------------------------------------------------------------
// GCNN residual layer (2x Chebyshev graph conv, K=5) for MI455X / gfx1250.
//   B=4, M=50000, F=64, K=5, NNZ=800000
// Phases per layer:
//   1) (layer1 only) transpose x[B,M,F] -> T0[M, F*B]        (51.2MB, L2-resident)
//   2) 4x SpMM scatter:  Y = -T_{k-2}; Y += 2*vals * gather  (atomics hit L2)
//   3) fused [200000x320]@[320x64] GEMM via V_WMMA_F32_16X16X4_F32 + ReLU
//      (+ residual add in layer 2)
// B fragments are staged in LDS row-pair interleaved so each fragment is one
// aligned ds_load_b64 straight into an even VGPR pair (no repack movs).
// ---------------------------------------------------------------------------

#define M_NODES 50000
#define B_BATCH 4
#define F_FEAT  64
#define K_CHEB  5
#define C_COLS  256                      // F*B
#define ROWS_TOT (B_BATCH * M_NODES)     // 200000
#define N_TILES  (ROWS_TOT / 16)         // 12500 (M % 16 == 0, tiles never cross b)
#define WAVES_PER_BLOCK 8
#define WL_ELEMS (K_CHEB * 32 * 64 * 2)  // 20480 floats = 80KB

typedef float v2f __attribute__((ext_vector_type(2)));
typedef float v8f __attribute__((ext_vector_type(8)));

// ----- T0[m*256 + f*4 + b] = x[b, m, f] ------------------------------------
__global__ void k_transpose(const float* __restrict__ x, float* __restrict__ T0) {
    int tid = blockIdx.x * blockDim.x + threadIdx.x;      // < M*256 = 12.8M
    if (tid >= M_NODES * C_COLS) return;
    int m = tid >> 8;
    int rem = tid & 255;
    int f = rem >> 2;
    int b = rem & 3;
    T0[tid] = x[(size_t)b * (M_NODES * F_FEAT) + (size_t)m * F_FEAT + f];
}

// ----- Y = -Tm2 (or 0) : folds the "- x_{k-2}" term into the accumulator ---
__global__ void k_init(float* __restrict__ Y, const float* __restrict__ Tm2, int n4) {
    int t = blockIdx.x * blockDim.x + threadIdx.x;
    if (t >= n4) return;
    float4* yp = ((float4*)Y) + t;
    if (Tm2) {
        float4 v = ((const float4*)Tm2)[t];
        *yp = make_float4(-v.x, -v.y, -v.z, -v.w);
    } else {
        *yp = make_float4(0.f, 0.f, 0.f, 0.f);
    }
}

// ----- Y[rows[e], :] += scale*vals[e] * X[cols[e], :] (scale = 1 or 2) -----
// 64 threads per edge, float4 per thread; [M,256] operand + accumulator both
// fit in the 192MB L2, so the atomics are L2-local.
__global__ void k_scatter(const int* __restrict__ rows, const int* __restrict__ cols,
                          const float* __restrict__ vals,
                          const float* __restrict__ X, float* __restrict__ Y,
                          float scale, int nnz) {
    long t = (long)blockIdx.x * blockDim.x + threadIdx.x;
    int e = (int)(t >> 6);
    if (e >= nnz) return;
    int c = ((int)t & 63) << 2;
    float v = vals[e] * scale;
    const float4 xv = *(const float4*)(X + (size_t)cols[e] * C_COLS + c);
    float* yp = Y + (size_t)rows[e] * C_COLS + c;
    atomicAdd(yp + 0, v * xv.x);
    atomicAdd(yp + 1, v * xv.y);
    atomicAdd(yp + 2, v * xv.z);
    atomicAdd(yp + 3, v * xv.w);
}

// ----- fused Chebyshev GEMM:  out[b*M+m, fo] = sum_{k,fin} T_k[m, fin*4+b] *
//       W[fin*5+k, fo]   via V_WMMA_F32_16X16X4_F32, one 16x64 tile per wave.
// mode 0: outH[m*256 + fo*4 + b] = relu(val)          (feeds layer-2 SpMM)
// mode 1: out[b,m,fo]            = relu(val + x[b,m,fo])
__global__ __launch_bounds__(256) void k_cheb_gemm(
    const float* __restrict__ T0, const float* __restrict__ T1,
    const float* __restrict__ T2, const float* __restrict__ T3,
    const float* __restrict__ T4,
    const float* __restrict__ W,        // [K*F, F], row = fin*K + k
    float* __restrict__ outp,
    const float* __restrict__ resid,    // mode 1: original x
    int mode) {
    // LDS layout: Wl[k][jp][n][p] = W[(2*jp+p)*K + k][n]
    //   k: cheb order (0..4), jp: K-row pair (0..31), n: out col (0..63), p: 0/1
    // -> a lane's B fragment {row kc, row kc+1} at col n is one aligned b64.
    extern __shared__ float Wl[];       // 20480 floats = 80KB

    for (int i = threadIdx.x; i < WL_ELEMS; i += blockDim.x) {
        int p  = i & 1;
        int n  = (i >> 1) & 63;
        int jp = (i >> 7) & 31;
        int k  = i >> 12;
        Wl[i] = W[((2 * jp + p) * K_CHEB + k) * F_FEAT + n];
    }
    __syncthreads();

    int lane = threadIdx.x & 31;
    int wave = threadIdx.x >> 5;
    int tile = blockIdx.x * WAVES_PER_BLOCK + wave;
    if (tile >= N_TILES) return;        // wave-uniform: EXEC all-ones for WMMA

    long bm0 = (long)tile * 16;
    int b  = (int)(bm0 / M_NODES);      // whole tile shares one batch index
    int m0 = (int)(bm0 % M_NODES);

    int half = lane >> 4;               // A frag: lanes 0-15 K{0,1}, 16-31 K{2,3}
    int l16  = lane & 15;
    int arow = m0 + l16;                // A-matrix row (M dim)

    const float* Tk[K_CHEB] = {T0, T1, T2, T3, T4};

    v8f acc0 = {}; v8f acc1 = {}; v8f acc2 = {}; v8f acc3 = {};

#pragma unroll
    for (int k = 0; k < K_CHEB; ++k) {
        const float* arowp = Tk[k] + (size_t)arow * C_COLS + b;
        const float* wk    = Wl + k * (32 * 64 * 2);   // 4096 floats per order
        for (int fin = 0; fin < F_FEAT; fin += 4) {
            int kc = fin + half * 2;                   // this lane's K pair
            v2f a;
            a.x = arowp[(kc + 0) * 4];                 // T_k[m, (kc  )*B + b]
            a.y = arowp[(kc + 1) * 4];                 // T_k[m, (kc+1)*B + b]
            // B frags: [jp = kc/2][col][2], row stride 128 floats, ntile +32
            const float* wp = wk + ((fin >> 1) + half) * 128 + l16 * 2;
            v2f b0 = *(const v2f*)(wp);
            v2f b1 = *(const v2f*)(wp + 32);
            v2f b2 = *(const v2f*)(wp + 64);
            v2f b3 = *(const v2f*)(wp + 96);
            acc0 = __builtin_amdgcn_wmma_f32_16x16x4_f32(false, a, false, b0, (short)0, acc0, false, false);
            acc1 = __builtin_amdgcn_wmma_f32_16x16x4_f32(false, a, false, b1, (short)0, acc1, false, false);
            acc2 = __builtin_amdgcn_wmma_f32_16x16x4_f32(false, a, false, b2, (short)0, acc2, false, false);
            acc3 = __builtin_amdgcn_wmma_f32_16x16x4_f32(false, a, false, b3, (short)0, acc3, false, false);
        }
    }

    // C/D layout: VGPR r -> M = r + 8*half, N = lane&15 (per 16x16 f32 table)
    v8f accs[4] = {acc0, acc1, acc2, acc3};
#pragma unroll
    for (int nt = 0; nt < 4; ++nt) {
        int col = nt * 16 + l16;
#pragma unroll
        for (int r = 0; r < 8; ++r) {
            float v = accs[nt][r];
            int m = m0 + r + half * 8;
            if (mode == 0) {
                outp[(size_t)m * C_COLS + col * 4 + b] = v > 0.f ? v : 0.f;
            } else {
                size_t idx = (size_t)b * (M_NODES * F_FEAT) + (size_t)m * F_FEAT + col;
                float o = v + resid[idx];
                outp[idx] = o > 0.f ? o : 0.f;
            }
        }
    }
}

// ---------------------------------------------------------------------------
extern "C" void kernel_launch(void* const* d_in, const int* in_sizes, int n_in,
                              void* d_out, int out_size, void* d_ws, size_t ws_size,
                              hipStream_t stream) {
    const float* x    = (const float*)d_in[0];
    const int*   rows = (const int*)d_in[1];
    const int*   cols = (const int*)d_in[2];
    const float* vals = (const float*)d_in[3];
    const float* W1   = (const float*)d_in[4];
    const float* W2   = (const float*)d_in[5];
    float* out = (float*)d_out;
    int nnz = in_sizes[1];

    const size_t S = (size_t)M_NODES * C_COLS;   // 12.8M floats per buffer
    float* ws = (float*)d_ws;
    float* T0 = ws;
    float* T1 = ws + S;
    float* T2 = ws + 2 * S;
    float* T3 = ws + 3 * S;
    float* T4 = ws + 4 * S;
    float* H  = ws + 5 * S;                      // layer-1 output in [M,256] layout

    const int thr = 256;
    int nElem = M_NODES * C_COLS;
    int n4 = nElem / 4;
    dim3 gT((nElem + thr - 1) / thr);
    dim3 gI((n4 + thr - 1) / thr);
    long scatThreads = (long)nnz * 64;
    dim3 gS((unsigned)((scatThreads + thr - 1) / thr));
    dim3 gG((N_TILES + WAVES_PER_BLOCK - 1) / WAVES_PER_BLOCK);
    size_t lds = (size_t)WL_ELEMS * sizeof(float);   // 80KB

    // -------- Layer 1 --------
    k_transpose<<<gT, thr, 0, stream>>>(x, T0);
    k_init   <<<gI, thr, 0, stream>>>(T1, nullptr, n4);
    k_scatter<<<gS, thr, 0, stream>>>(rows, cols, vals, T0, T1, 1.0f, nnz);
    k_init   <<<gI, thr, 0, stream>>>(T2, T0, n4);
    k_scatter<<<gS, thr, 0, stream>>>(rows, cols, vals, T1, T2, 2.0f, nnz);
    k_init   <<<gI, thr, 0, stream>>>(T3, T1, n4);
    k_scatter<<<gS, thr, 0, stream>>>(rows, cols, vals, T2, T3, 2.0f, nnz);
    k_init   <<<gI, thr, 0, stream>>>(T4, T2, n4);
    k_scatter<<<gS, thr, 0, stream>>>(rows, cols, vals, T3, T4, 2.0f, nnz);
    k_cheb_gemm<<<gG, thr, lds, stream>>>(T0, T1, T2, T3, T4, W1, H, nullptr, 0);

    // -------- Layer 2 (input = H, reuse T1..T4) --------
    k_init   <<<gI, thr, 0, stream>>>(T1, nullptr, n4);
    k_scatter<<<gS, thr, 0, stream>>>(rows, cols, vals, H, T1, 1.0f, nnz);
    k_init   <<<gI, thr, 0, stream>>>(T2, H, n4);
    k_scatter<<<gS, thr, 0, stream>>>(rows, cols, vals, T1, T2, 2.0f, nnz);
    k_init   <<<gI, thr, 0, stream>>>(T3, T1, n4);
    k_scatter<<<gS, thr, 0, stream>>>(rows, cols, vals, T2, T3, 2.0f, nnz);
    k_init   <<<gI, thr, 0, stream>>>(T4, T2, n4);
    k_scatter<<<gS, thr, 0, stream>>>(rows, cols, vals, T3, T4, 2.0f, nnz);
    k_cheb_gemm<<<gG, thr, lds, stream>>>(H, T1, T2, T3, T4, W2, out, x, 1);
}